// GIN_11751030522723
// MI455X (gfx1250) — compile-verified
//
#include <hip/hip_runtime.h>

typedef __attribute__((ext_vector_type(2))) float v2f;
typedef __attribute__((ext_vector_type(8))) float v8f;

#define F_IN 128
#define HID  256
#define NCLS 40
#define NCLS_PAD 64

// ---------------- simple float4 copy ----------------
__global__ void copy_f4(const float4* __restrict__ in, float4* __restrict__ out, int count4) {
  int i = blockIdx.x * blockDim.x + threadIdx.x;
  if (i < count4) out[i] = in[i];
}

// ---------------- transpose W[k][Hout] -> Wt[n][K], zero-pad rows n >= Hout ----
__global__ void transpose_w(const float* __restrict__ W, float* __restrict__ Wt,
                            int K, int Hout, int HoutPad) {
  int i = blockIdx.x * blockDim.x + threadIdx.x;
  if (i >= K * HoutPad) return;
  int n = i / K;
  int k = i - n * K;
  Wt[i] = (n < Hout) ? W[k * Hout + n] : 0.0f;
}

// ---------------- edge scatter: t[dst] += h[src], one wave32 per edge ----------
template<int D>
__global__ void scatter_add(float* __restrict__ t, const float* __restrict__ h,
                            const int* __restrict__ src, const int* __restrict__ dst,
                            int E) {
  int wid  = (blockIdx.x * blockDim.x + threadIdx.x) >> 5;  // wave32
  int lane = threadIdx.x & 31;
  if (wid >= E) return;  // uniform per wave
  size_t s = (size_t)src[wid] * D;
  size_t d = (size_t)dst[wid] * D;
  #pragma unroll
  for (int i = 0; i < D / 128; ++i) {
    int c = (lane + 32 * i) * 4;
    const float4 v = *(const float4*)(h + s + c);
    float* p = t + d + c;
    __hip_atomic_fetch_add(p + 0, v.x, __ATOMIC_RELAXED, __HIP_MEMORY_SCOPE_AGENT);
    __hip_atomic_fetch_add(p + 1, v.y, __ATOMIC_RELAXED, __HIP_MEMORY_SCOPE_AGENT);
    __hip_atomic_fetch_add(p + 2, v.z, __ATOMIC_RELAXED, __HIP_MEMORY_SCOPE_AGENT);
    __hip_atomic_fetch_add(p + 3, v.w, __ATOMIC_RELAXED, __HIP_MEMORY_SCOPE_AGENT);
  }
}

// ---------------- WMMA f32 GEMM: out = act(T[N,K] @ W[K,Hout] + bias) ----------
// Wt is W transposed: Wt[n*K + k] = W[k*Hout + n], with HoutPad rows (zero-padded).
// Block = 128 threads = 4 waves (2x2); each wave computes a 32x32 output tile.
template<int K, bool RELU>
__global__ void gemm_wmma(const float* __restrict__ T, const float* __restrict__ Wt,
                          const float* __restrict__ bias, float* __restrict__ out,
                          int Nrows, int Hout) {
  const int lane = threadIdx.x & 31;
  const int wave = threadIdx.x >> 5;         // 0..3
  const int wm = wave & 1, wn = wave >> 1;   // 2x2 wave grid
  const int m0 = blockIdx.x * 64 + wm * 32;
  const int n0 = blockIdx.y * 64 + wn * 32;
  const int l16 = lane & 15;
  const int kh  = lane >> 4;                 // K half-select within frag

  // A rows (clamped so loads are always in-bounds; clamped rows never stored)
  const int rowA = m0 + l16, rowB = m0 + 16 + l16;
  const int rA = rowA < Nrows ? rowA : Nrows - 1;
  const int rB = rowB < Nrows ? rowB : Nrows - 1;
  // B columns (Wt is padded to HoutPad rows, always in-bounds)
  const int nA = n0 + l16, nB = n0 + 16 + l16;

  const float* pA = T  + (size_t)rA * K + 2 * kh;
  const float* pB = T  + (size_t)rB * K + 2 * kh;
  const float* qA = Wt + (size_t)nA * K + 2 * kh;
  const float* qB = Wt + (size_t)nB * K + 2 * kh;

  v8f acc00 = {}, acc01 = {}, acc10 = {}, acc11 = {};
  #pragma unroll 8
  for (int k = 0; k < K; k += 4) {
    // f32 16x4 A-frag: a[v] = A(l16, k + v + 2*kh)  -> contiguous b64 load
    v2f a0 = *(const v2f*)(pA + k);
    v2f a1 = *(const v2f*)(pB + k);
    // f32 4x16 B-frag: b[v] = B(k + v + 2*kh, l16) = Wt[n][k+v+2*kh] -> contiguous
    v2f b0 = *(const v2f*)(qA + k);
    v2f b1 = *(const v2f*)(qB + k);
    acc00 = __builtin_amdgcn_wmma_f32_16x16x4_f32(false, a0, false, b0, (short)0, acc00, false, false);
    acc01 = __builtin_amdgcn_wmma_f32_16x16x4_f32(false, a0, false, b1, (short)0, acc01, false, false);
    acc10 = __builtin_amdgcn_wmma_f32_16x16x4_f32(false, a1, false, b0, (short)0, acc10, false, false);
    acc11 = __builtin_amdgcn_wmma_f32_16x16x4_f32(false, a1, false, b1, (short)0, acc11, false, false);
  }

  const float bA = bias[nA < Hout ? nA : Hout - 1];
  const float bB = bias[nB < Hout ? nB : Hout - 1];
  // C/D layout: c[v] = C(v + 8*kh, l16)
  #pragma unroll
  for (int v = 0; v < 8; ++v) {
    int r0 = m0 + v + 8 * kh;
    int r1 = r0 + 16;
    float e00 = acc00[v] + bA, e01 = acc01[v] + bB;
    float e10 = acc10[v] + bA, e11 = acc11[v] + bB;
    if (RELU) {
      e00 = fmaxf(e00, 0.f); e01 = fmaxf(e01, 0.f);
      e10 = fmaxf(e10, 0.f); e11 = fmaxf(e11, 0.f);
    }
    if (r0 < Nrows) {
      if (nA < Hout) out[(size_t)r0 * Hout + nA] = e00;
      if (nB < Hout) out[(size_t)r0 * Hout + nB] = e01;
    }
    if (r1 < Nrows) {
      if (nA < Hout) out[(size_t)r1 * Hout + nA] = e10;
      if (nB < Hout) out[(size_t)r1 * Hout + nB] = e11;
    }
  }
}

extern "C" void kernel_launch(void* const* d_in, const int* in_sizes, int n_in,
                              void* d_out, int out_size, void* d_ws, size_t ws_size,
                              hipStream_t stream) {
  const float* x  = (const float*)d_in[0];
  const float* W1 = (const float*)d_in[1];
  const float* b1 = (const float*)d_in[2];
  const float* W2 = (const float*)d_in[3];
  const float* b2 = (const float*)d_in[4];
  const float* W3 = (const float*)d_in[5];
  const float* b3 = (const float*)d_in[6];
  const int*  src = (const int*)d_in[7];
  const int*  dst = (const int*)d_in[8];

  const int N = in_sizes[0] / F_IN;
  const int E = in_sizes[7];

  // Workspace layout (floats): bufA = N*HID (h1 then h2), bufB = N*HID (t1 then t2),
  // then the three transposed weight matrices. ~205 MB total.
  float* ws   = (float*)d_ws;
  size_t NH   = (size_t)N * HID;
  float* bufA = ws;
  float* bufB = ws + NH;
  float* wt1  = ws + 2 * NH;                 // [HID][F_IN]
  float* wt2  = wt1 + (size_t)HID * F_IN;    // [HID][HID]
  float* wt3  = wt2 + (size_t)HID * HID;     // [NCLS_PAD][HID]

  // Pre-transpose weights (tiny; independent of graph work).
  { int tot = HID * F_IN;      transpose_w<<<(tot + 255) / 256, 256, 0, stream>>>(W1, wt1, F_IN, HID, HID); }
  { int tot = HID * HID;       transpose_w<<<(tot + 255) / 256, 256, 0, stream>>>(W2, wt2, HID, HID, HID); }
  { int tot = NCLS_PAD * HID;  transpose_w<<<(tot + 255) / 256, 256, 0, stream>>>(W3, wt3, HID, NCLS, NCLS_PAD); }

  // Layer 1: t1 = x + segment_sum(x[src] -> dst); h1 = relu(t1 @ W1 + b1)
  { int c4 = (N * F_IN) / 4;  copy_f4<<<(c4 + 255) / 256, 256, 0, stream>>>((const float4*)x, (float4*)bufB, c4); }
  scatter_add<F_IN><<<(E + 7) / 8, 256, 0, stream>>>(bufB, x, src, dst, E);
  dim3 gH((N + 63) / 64, HID / 64);
  gemm_wmma<F_IN, true><<<gH, 128, 0, stream>>>(bufB, wt1, b1, bufA, N, HID);

  // Layer 2: t2 = h1 + segment_sum(h1[src] -> dst); h2 = relu(t2 @ W2 + b2)
  { int c4 = (int)(NH / 4);   copy_f4<<<(c4 + 255) / 256, 256, 0, stream>>>((const float4*)bufA, (float4*)bufB, c4); }
  scatter_add<HID><<<(E + 7) / 8, 256, 0, stream>>>(bufB, bufA, src, dst, E);
  gemm_wmma<HID, true><<<gH, 128, 0, stream>>>(bufB, wt2, b2, bufA, N, HID);

  // Head: out = h2 @ W3 + b3
  dim3 gC((N + 63) / 64, NCLS_PAD / 64);
  gemm_wmma<HID, false><<<gC, 128, 0, stream>>>(bufA, wt3, b3, (float*)d_out, N, NCLS);
}